// MusicTokenEnforcementLoss_31233002176528
// MI455X (gfx1250) — compile-verified
//
#include <hip/hip_runtime.h>
#include <hip/hip_bf16.h>
#include <float.h>

#define NEGV    (-1e10f)
#define MAXK    8
#define BS      256
#define PENALTY 100.0f
#define EPSV    1e-12f

typedef __attribute__((address_space(1))) const unsigned char* gcu8_t;
typedef __attribute__((address_space(3))) unsigned char*       lu8_t;

typedef int v4i_t __attribute__((ext_vector_type(4)));
typedef __attribute__((address_space(1))) v4i_t* gv4i_p;   // printed as "__device__ int4*"
typedef __attribute__((address_space(3))) v4i_t* lv4i_p;   // printed as "__shared__ int4*"

// Async global->LDS copy of one 16B chunk (CDNA5 path, ASYNCcnt-tracked).
__device__ __forceinline__ void async_copy16_to_lds(const unsigned char* gsrc,
                                                    unsigned char* ldst) {
#if __has_builtin(__builtin_amdgcn_global_load_async_to_lds_b128)
    __builtin_amdgcn_global_load_async_to_lds_b128(
        (gv4i_p)(gcu8_t)gsrc, (lv4i_p)(lu8_t)ldst, 0, 0);
#else
    unsigned lofs = (unsigned)(unsigned long long)(lu8_t)ldst;
    asm volatile("global_load_async_to_lds_b128 %0, %1, off"
                 :: "v"(lofs), "v"(gsrc) : "memory");
#endif
}

__device__ __forceinline__ void wait_asynccnt0() {
#if __has_builtin(__builtin_amdgcn_s_wait_asynccnt)
    __builtin_amdgcn_s_wait_asynccnt(0);
#else
    asm volatile("s_wait_asynccnt 0x0" ::: "memory");
#endif
}

// One block (256 threads = 8 wave32) per (b,s) row. Single streaming pass over
// V logits: online masked logsumexp + per-thread top-K list, then wave32
// shuffle + LDS merges. The non-music mask is staged into LDS once per block
// via async direct-to-LDS loads; thread 0 finishes CE/NLL and the penalty.
__global__ __launch_bounds__(BS) void mtel_row_kernel(
    const float* __restrict__ logits,
    const int* __restrict__ labels,
    const int* __restrict__ attn,
    const unsigned char* __restrict__ nm,   // bool mask, 1 byte/elem
    const int* __restrict__ topk_ptr,
    float* __restrict__ ws_nll,
    float* __restrict__ ws_keep,
    float* __restrict__ ws_pen,
    float* __restrict__ ws_flag,
    int V)
{
    const int row = blockIdx.x;
    const int tid = threadIdx.x;

    __shared__ float red_v[BS];
    __shared__ int   red_i[BS];
    __shared__ int   red_t[BS];
    __shared__ float wm_s[BS / 32];
    __shared__ float ws_s[BS / 32];
    __shared__ float winv[MAXK];
    __shared__ int   wini[MAXK];
    extern __shared__ unsigned char smem_nm[];   // V bytes (dynamic LDS)

    // ---- stage non_music_mask into LDS (async direct-to-LDS, 16B chunks) ----
    {
        const int nVec = V >> 4;                 // # of full 16-byte chunks
        for (int i = tid; i < nVec; i += BS)
            async_copy16_to_lds(nm + ((size_t)i << 4), smem_nm + ((size_t)i << 4));
        for (int j = (nVec << 4) + tid; j < V; j += BS)
            smem_nm[j] = nm[j];                  // tail bytes via DS store
        wait_asynccnt0();
    }
    __syncthreads();

    int K = topk_ptr[0];
    K = K < 1 ? 1 : (K > MAXK ? MAXK : K);

    const int  label = labels[row];
    const bool keep  = (label != -100);
    const bool valid = keep && (attn[row] == 1);

    const float* rowp = logits + (size_t)row * (size_t)V;

    // thread-local top-K (sorted descending), kept in registers via full unroll
    float tv[MAXK];
    int   ti[MAXK];
#pragma unroll
    for (int k = 0; k < MAXK; ++k) { tv[k] = -FLT_MAX; ti[k] = 0x7fffffff; }
    float kmin = -FLT_MAX;

    // online logsumexp state (over masked logits)
    float m = -FLT_MAX, ssum = 0.0f;

    for (int j = tid; j < V; j += BS) {
        float v = rowp[j];
        int pf = j + BS * 8;
        if (pf < V) __builtin_prefetch(rowp + pf, 0, 1);  // global_prefetch_b8
        const bool nmb = (smem_nm[j] != 0);               // LDS byte read

        // ---- top-K insertion (strict >, preserves lower-index tie order) ----
        if (v > kmin) {
            bool gt[MAXK];
#pragma unroll
            for (int k = 0; k < MAXK; ++k) gt[k] = (k < K) && (v > tv[k]);
#pragma unroll
            for (int k = MAXK - 1; k >= 1; --k) {
                if (gt[k - 1]) { tv[k] = tv[k - 1]; ti[k] = ti[k - 1]; }
            }
#pragma unroll
            for (int k = 0; k < MAXK; ++k) {
                if (gt[k] && (k == 0 || !gt[k - 1])) { tv[k] = v; ti[k] = j; }
            }
#pragma unroll
            for (int k = 0; k < MAXK; ++k) if (k == K - 1) kmin = tv[k];
        }

        // ---- online masked logsumexp: skip masked-out (-> -1e10) elements ----
        if (!(valid && nmb)) {
            if (v > m) { ssum = ssum * __expf(m - v) + 1.0f; m = v; }
            else       { ssum += __expf(v - m); }
        }
    }

    // ---- wave32 reduction of (m, ssum) ----
#pragma unroll
    for (int off = 16; off > 0; off >>= 1) {
        float m2 = __shfl_xor(m, off, 32);
        float s2 = __shfl_xor(ssum, off, 32);
        float M  = fmaxf(m, m2);
        ssum = ssum * __expf(m - M) + s2 * __expf(m2 - M);
        m = M;
    }
    const int wid  = tid >> 5;
    const int lane = tid & 31;
    if (lane == 0) { wm_s[wid] = m; ws_s[wid] = ssum; }
    __syncthreads();

    // ---- block-wide top-K merge: K rounds of argmax over per-thread heads ----
    int head = 0;
    for (int r = 0; r < K; ++r) {
        float cv = -FLT_MAX;
        int   ci = 0x7fffffff;
#pragma unroll
        for (int k = 0; k < MAXK; ++k)
            if (k == head && k < K) { cv = tv[k]; ci = ti[k]; }

        red_v[tid] = cv; red_i[tid] = ci; red_t[tid] = tid;
        __syncthreads();
        for (int off = BS / 2; off > 0; off >>= 1) {
            if (tid < off) {
                float ov = red_v[tid + off];
                int   oi = red_i[tid + off];
                int   ot = red_t[tid + off];
                float mv = red_v[tid];
                int   mi = red_i[tid];
                bool take = (ov > mv) || (ov == mv && oi < mi);
                if (take) { red_v[tid] = ov; red_i[tid] = oi; red_t[tid] = ot; }
            }
            __syncthreads();
        }
        if (tid == 0) { winv[r] = red_v[0]; wini[r] = red_i[0]; }
        int wt = red_t[0];
        __syncthreads();
        if (wt == tid) head++;
    }

    // ---- thread 0: finalize this row ----
    if (tid == 0) {
        const int nw = BS / 32;
        float M = wm_s[0], S = ws_s[0];
        for (int w = 1; w < nw; ++w) {
            float m2 = wm_s[w], s2 = ws_s[w];
            float Mn = fmaxf(M, m2);
            S = S * __expf(M - Mn) + s2 * __expf(m2 - Mn);
            M = Mn;
        }
        float lse = M + logf(S);

        float nll = 0.0f;
        if (keep) {
            float lv = rowp[label];
            float ml = (valid && smem_nm[label] != 0) ? NEGV : lv;
            nll = lse - ml;
        }

        // top-k penalty on ORIGINAL logits
        float mx = winv[0];
        float sum = 0.0f, pmax = 0.0f;
        bool any = false;
        for (int k = 0; k < K; ++k) {
            float e = expf(winv[k] - mx);
            sum += e;
            if (smem_nm[wini[k]] != 0) { any = true; pmax = fmaxf(pmax, e); }
        }
        float pen = 0.0f, flag = 0.0f;
        if (any && valid) {
            float p = pmax / sum;
            p = fmaxf(p, EPSV);
            pen = -logf(p) * PENALTY;
            flag = 1.0f;
        }

        ws_nll[row]  = keep ? nll : 0.0f;
        ws_keep[row] = keep ? 1.0f : 0.0f;
        ws_pen[row]  = pen;
        ws_flag[row] = flag;
    }
}

// Deterministic fixed-geometry final reduction over R rows -> 3 scalars.
__global__ __launch_bounds__(BS) void mtel_finalize(
    const float* __restrict__ ws_nll,
    const float* __restrict__ ws_keep,
    const float* __restrict__ ws_pen,
    const float* __restrict__ ws_flag,
    float* __restrict__ out,
    int R)
{
    __shared__ float s0[BS], s1[BS], s2[BS], s3[BS];
    const int tid = threadIdx.x;
    float a = 0.f, b = 0.f, c = 0.f, d = 0.f;
    for (int i = tid; i < R; i += BS) {
        a += ws_nll[i]; b += ws_keep[i]; c += ws_pen[i]; d += ws_flag[i];
    }
    s0[tid] = a; s1[tid] = b; s2[tid] = c; s3[tid] = d;
    __syncthreads();
    for (int off = BS / 2; off > 0; off >>= 1) {
        if (tid < off) {
            s0[tid] += s0[tid + off];
            s1[tid] += s1[tid + off];
            s2[tid] += s2[tid + off];
            s3[tid] += s3[tid + off];
        }
        __syncthreads();
    }
    if (tid == 0) {
        float nk  = s1[0] < 1.0f ? 1.0f : s1[0];
        float ce  = s0[0] / nk;
        float np  = s3[0];
        float pen = (np > 0.0f) ? (s2[0] / (np < 1.0f ? 1.0f : np)) : 0.0f;
        out[0] = ce + pen;  // total_loss
        out[1] = ce;        // ce_loss
        out[2] = pen;       // penalty_loss
    }
}

extern "C" void kernel_launch(void* const* d_in, const int* in_sizes, int n_in,
                              void* d_out, int out_size, void* d_ws, size_t ws_size,
                              hipStream_t stream) {
    const float*         logits = (const float*)d_in[0];
    const int*           labels = (const int*)d_in[1];
    const int*           attn   = (const int*)d_in[2];
    const unsigned char* nm     = (const unsigned char*)d_in[3]; // bool -> 1 byte
    const int*           topk   = (const int*)d_in[4];

    const int R = in_sizes[1];  // B*S rows
    const int V = in_sizes[3];  // vocab

    float* ws      = (float*)d_ws;
    float* ws_nll  = ws;
    float* ws_keep = ws + (size_t)R;
    float* ws_pen  = ws + (size_t)2 * R;
    float* ws_flag = ws + (size_t)3 * R;

    const size_t smbytes = ((size_t)V + 15) & ~(size_t)15;  // dynamic LDS for mask

    mtel_row_kernel<<<R, BS, smbytes, stream>>>(logits, labels, attn, nm, topk,
                                                ws_nll, ws_keep, ws_pen, ws_flag, V);
    mtel_finalize<<<1, BS, 0, stream>>>(ws_nll, ws_keep, ws_pen, ws_flag,
                                        (float*)d_out, R);
}